// CondRealNVPFlow3D_64931315581406
// MI455X (gfx1250) — compile-verified
//
#include <hip/hip_runtime.h>
#include <math.h>

#define B_   64
#define N_   16384
#define G_   512
#define F_   64
#define M_   (B_ * N_)          // 1048576 columns
#define OUTS 3145728            // 64*3*16384 per output tensor

typedef float v2f __attribute__((ext_vector_type(2)));
typedef float v8f __attribute__((ext_vector_type(8)));

// ---- workspace layout (float offsets) ----
#define NWG0 256
#define NWG2 256
#define OFF_MOMP  0                          // NWG0*5 moment partials
#define OFF_BN1   (NWG0 * 5)                 // 2 branches * (A0,A1,C)[64] = 384
#define OFF_GRAM  (OFF_BN1 + 2 * 192)        // reduced gram 64*64 + colsum 64 = 4160 (reused per branch)
#define OFF_W2    (OFF_GRAM + 4160)          // 2 branches * (W' 4096 + d 64)
#define OFF_FILM  (OFF_W2 + 2 * 4160)        // 2 branches * (scale 4096 + bias 4096)
#define OFF_GRAMP (OFF_FILM + 2 * 8192)      // NWG2 * 4160 gram partials (~4.3 MB total ws)

// ---------------- K0: global moments of p1,p2 ----------------
__global__ void k0_moments(const float* __restrict__ p, float* __restrict__ ws) {
    __shared__ float red[256];
    float s1 = 0.f, s2 = 0.f, s11 = 0.f, s22 = 0.f, s12 = 0.f;
    int gid = blockIdx.x * 256 + threadIdx.x;
    for (int i = gid; i < M_; i += NWG0 * 256) {
        int b = i >> 14, n = i & (N_ - 1);
        const float* base = p + (long)b * 3 * N_ + n;
        float a = base[N_], c = base[2 * N_];
        s1 += a; s2 += c; s11 += a * a; s22 += c * c; s12 += a * c;
    }
    float vals[5] = { s1, s2, s11, s22, s12 };
    for (int j = 0; j < 5; ++j) {
        red[threadIdx.x] = vals[j];
        __syncthreads();
        for (int off = 128; off > 0; off >>= 1) {
            if (threadIdx.x < off) red[threadIdx.x] += red[threadIdx.x + off];
            __syncthreads();
        }
        if (threadIdx.x == 0) ws[OFF_MOMP + blockIdx.x * 5 + j] = red[0];
        __syncthreads();
    }
}

// ---------------- K1: reduce moments, fold BN1 (both branches) ----------------
__global__ void k1_prep(float* __restrict__ ws,
                        const float* __restrict__ lvW, const float* __restrict__ lvG, const float* __restrict__ lvB,
                        const float* __restrict__ muW, const float* __restrict__ muG, const float* __restrict__ muB) {
    __shared__ float mom[5];
    if (threadIdx.x < 5) {
        float s = 0.f;
        for (int w = 0; w < NWG0; ++w) s += ws[OFF_MOMP + w * 5 + threadIdx.x];
        mom[threadIdx.x] = s;
    }
    __syncthreads();
    const float inv = 1.0f / (float)M_;
    for (int idx = threadIdx.x; idx < 128; idx += blockDim.x) {
        int br = idx >> 6, c = idx & 63;
        const float* W  = br ? muW : lvW;
        const float* Gm = br ? muG : lvG;
        const float* Bt = br ? muB : lvB;
        float w0 = W[c * 2 + 0], w1 = W[c * 2 + 1];
        float mean = (w0 * mom[0] + w1 * mom[1]) * inv;
        float e2   = (w0 * w0 * mom[2] + 2.f * w0 * w1 * mom[4] + w1 * w1 * mom[3]) * inv;
        float var  = e2 - mean * mean;
        float a    = Gm[c] * rsqrtf(var + 1e-5f);
        ws[OFF_BN1 + br * 192 +   0 + c] = a * w0;
        ws[OFF_BN1 + br * 192 +  64 + c] = a * w1;
        ws[OFF_BN1 + br * 192 + 128 + c] = Bt[c] - a * mean;
    }
}

// ---------------- K2: Gram matrix of h1 via WMMA f32 16x16x4 ----------------
__global__ void __launch_bounds__(256) k2_gram(const float* __restrict__ p,
                                               float* __restrict__ ws, int br) {
    __shared__ float H[64][65];
    __shared__ float A0[64], A1[64], C0[64];
    int t = threadIdx.x;
    if (t < 64) {
        A0[t] = ws[OFF_BN1 + br * 192 + t];
        A1[t] = ws[OFF_BN1 + br * 192 + 64 + t];
        C0[t] = ws[OFF_BN1 + br * 192 + 128 + t];
    }
    __syncthreads();

    int lane = t & 31, wave = t >> 5;
    int lo = lane & 15, hi = lane >> 4;
    int tile0 = wave * 2;
    int i0 = tile0 >> 2, j0 = tile0 & 3, j1 = (tile0 + 1) & 3;   // tile pair shares i
    v8f acc0 = {0.f,0.f,0.f,0.f,0.f,0.f,0.f,0.f};
    v8f acc1 = acc0;

    int ct = t & 63, gt = t >> 6;
    float ssum[16];
    #pragma unroll
    for (int q = 0; q < 16; ++q) ssum[q] = 0.f;

    // 64 chunks of 64 columns; each WG spans 4096 cols inside one batch
    for (int cc = 0; cc < 64; ++cc) {
        long col = (long)blockIdx.x * 4096 + cc * 64;
        int b = (int)(col >> 14);
        int n = (int)(col & (N_ - 1)) + ct;
        float p1v = p[(long)b * 3 * N_ + N_ + n];
        float p2v = p[(long)b * 3 * N_ + 2 * N_ + n];
        #pragma unroll
        for (int q = 0; q < 16; ++q) {
            int ch = gt * 16 + q;
            float h = fmaxf(A0[ch] * p1v + A1[ch] * p2v + C0[ch], 0.f);
            H[ch][ct] = h;
            ssum[q] += h;
        }
        __syncthreads();
        #pragma unroll
        for (int kk = 0; kk < 64; kk += 4) {
            v2f a, b0, b1;
            a.x  = H[i0 * 16 + lo][kk + hi * 2];      a.y  = H[i0 * 16 + lo][kk + hi * 2 + 1];
            b0.x = H[j0 * 16 + lo][kk + hi * 2];      b0.y = H[j0 * 16 + lo][kk + hi * 2 + 1];
            b1.x = H[j1 * 16 + lo][kk + hi * 2];      b1.y = H[j1 * 16 + lo][kk + hi * 2 + 1];
            acc0 = __builtin_amdgcn_wmma_f32_16x16x4_f32(false, a, false, b0, (short)0, acc0, false, false);
            acc1 = __builtin_amdgcn_wmma_f32_16x16x4_f32(false, a, false, b1, (short)0, acc1, false, false);
        }
        __syncthreads();
    }

    float* gp = ws + OFF_GRAMP + (long)blockIdx.x * 4160;
    #pragma unroll
    for (int r = 0; r < 8; ++r) {
        int gm = i0 * 16 + r + hi * 8;
        gp[gm * 64 + j0 * 16 + lo] = acc0[r];
        gp[gm * 64 + j1 * 16 + lo] = acc1[r];
    }
    // column sums (reuse H as staging)
    #pragma unroll
    for (int q = 0; q < 16; ++q) H[gt * 16 + q][ct] = ssum[q];
    __syncthreads();
    if (t < 64) {
        float s = 0.f;
        for (int c2 = 0; c2 < 64; ++c2) s += H[t][c2];
        gp[4096 + t] = s;
    }
}

// ---------------- K2b: reduce gram partials ----------------
__global__ void k2b_reduce(float* __restrict__ ws) {
    int e = blockIdx.x * 256 + threadIdx.x;
    if (e >= 4160) return;
    float s = 0.f;
    for (int w = 0; w < NWG2; ++w) s += ws[OFF_GRAMP + (long)w * 4160 + e];
    ws[OFF_GRAM + e] = s;
}

// ---------------- K2c: fold BN2 into W' and d ----------------
__global__ void k2c_bn2(float* __restrict__ ws, const float* __restrict__ W, int br) {
    int c = threadIdx.x;
    if (c >= 64) return;
    const float* Gm = ws + OFF_GRAM;
    const float* sv = ws + OFF_GRAM + 4096;
    const float inv = 1.0f / (float)M_;
    float mean = 0.f;
    for (int k = 0; k < 64; ++k) mean += W[c * 64 + k] * sv[k];
    mean *= inv;
    float e2 = 0.f;
    for (int k = 0; k < 64; ++k) {
        float acc = 0.f;
        for (int l = 0; l < 64; ++l) acc += Gm[k * 64 + l] * W[c * 64 + l];
        e2 += W[c * 64 + k] * acc;
    }
    e2 *= inv;
    float var = e2 - mean * mean;
    float rs  = rsqrtf(var + 1e-5f);
    float* Wp = ws + OFF_W2 + br * 4160;
    for (int k = 0; k < 64; ++k) Wp[c * 64 + k] = rs * W[c * 64 + k];
    Wp[4096 + c] = -rs * mean;
}

// ---------------- K3: FiLM net (tiny) ----------------
__global__ void k3_film(const float* __restrict__ g, const float* __restrict__ W0,
                        const float* __restrict__ bng, const float* __restrict__ bnb,
                        const float* __restrict__ W1, const float* __restrict__ b1,
                        float* __restrict__ outp, int is_scale) {
    __shared__ float H[64][65];
    int t = threadIdx.x;
    for (int o = t; o < 4096; o += 256) {
        int b = o >> 6, c = o & 63;
        float acc = 0.f;
        for (int k = 0; k < G_; ++k) acc += g[b * G_ + k] * W0[c * G_ + k];
        H[b][c] = acc;
    }
    __syncthreads();
    if (t < 64) {                       // BN over batch axis + SiLU, one column per thread
        float m = 0.f;
        for (int b = 0; b < 64; ++b) m += H[b][t];
        m *= (1.f / 64.f);
        float v = 0.f;
        for (int b = 0; b < 64; ++b) { float d = H[b][t] - m; v += d * d; }
        v *= (1.f / 64.f);
        float rs = rsqrtf(v + 1e-5f);
        float gg = bng[t], bb = bnb[t];
        for (int b = 0; b < 64; ++b) {
            float x = (H[b][t] - m) * rs * gg + bb;
            H[b][t] = x / (1.f + expf(-x));        // x * sigmoid(x)
        }
    }
    __syncthreads();
    for (int o = t; o < 4096; o += 256) {
        int b = o >> 6, c = o & 63;
        float acc = b1[c];
        for (int k = 0; k < 64; ++k) acc += H[b][k] * W1[c * 64 + k];
        outp[o] = is_scale ? (1e-6f + expf(acc)) : acc;
    }
}

// ---------------- K4: fused main pass (WMMA GEMM + FiLM + relu + sd2 reduce) ----------------
__global__ void __launch_bounds__(256) k4_main(const float* __restrict__ p,
                                               const float* __restrict__ ws,
                                               const float* __restrict__ V64,
                                               const float* __restrict__ sb1,
                                               float* __restrict__ outbuf, int br) {
    __shared__ float Wp[64][65];
    __shared__ float h1[64][65];
    __shared__ float A0[64], A1[64], C0[64], D0[64], SC[64], TB[64], VV[64];
    __shared__ float part[4][64];
    int t = threadIdx.x;
    int wg = blockIdx.x;
    int b = wg >> 5;                       // 512 cols per WG, 32 WGs per batch
    const float* wpsrc = ws + OFF_W2 + br * 4160;
    for (int o = t; o < 4096; o += 256) Wp[o >> 6][o & 63] = wpsrc[o];
    if (t < 64) {
        A0[t] = ws[OFF_BN1 + br * 192 + t];
        A1[t] = ws[OFF_BN1 + br * 192 + 64 + t];
        C0[t] = ws[OFF_BN1 + br * 192 + 128 + t];
        D0[t] = wpsrc[4096 + t];
        SC[t] = ws[OFF_FILM + br * 8192 + b * 64 + t];
        TB[t] = ws[OFF_FILM + br * 8192 + 4096 + b * 64 + t];
        VV[t] = V64[t];
    }
    __syncthreads();
    float sb = sb1[0];

    int lane = t & 31, wave = t >> 5;
    int lo = lane & 15, hi = lane >> 4;
    int tile0 = wave * 2;
    int i0 = tile0 >> 2, j0 = tile0 & 3, j1 = (tile0 + 1) & 3;
    int ct = t & 63, gt = t >> 6;
    int nbase = (int)(((long)wg << 9) & (N_ - 1));    // in-batch column base
    const float* p1p = p + (long)b * 3 * N_ + N_;
    const float* p2p = p + (long)b * 3 * N_ + 2 * N_;
    // lv (br==0) -> logvar slot (3rd output); mu (br==1) -> mu slot (2nd output)
    float* orow = outbuf + (br == 0 ? (long)2 * OUTS : (long)OUTS) + (long)b * 3 * N_;

    for (int cc = 0; cc < 8; ++cc) {
        int n0 = nbase + cc * 64;
        float p1v = p1p[n0 + ct];
        float p2v = p2p[n0 + ct];
        #pragma unroll
        for (int q = 0; q < 16; ++q) {
            int ch = gt * 16 + q;
            h1[ch][ct] = fmaxf(A0[ch] * p1v + A1[ch] * p2v + C0[ch], 0.f);
        }
        __syncthreads();

        v8f acc0 = {0.f,0.f,0.f,0.f,0.f,0.f,0.f,0.f};
        v8f acc1 = acc0;
        #pragma unroll
        for (int kk = 0; kk < 64; kk += 4) {
            v2f a, bb0, bb1;
            a.x   = Wp[i0 * 16 + lo][kk + hi * 2];    a.y   = Wp[i0 * 16 + lo][kk + hi * 2 + 1];
            bb0.x = h1[kk + hi * 2][j0 * 16 + lo];    bb0.y = h1[kk + hi * 2 + 1][j0 * 16 + lo];
            bb1.x = h1[kk + hi * 2][j1 * 16 + lo];    bb1.y = h1[kk + hi * 2 + 1][j1 * 16 + lo];
            acc0 = __builtin_amdgcn_wmma_f32_16x16x4_f32(false, a, false, bb0, (short)0, acc0, false, false);
            acc1 = __builtin_amdgcn_wmma_f32_16x16x4_f32(false, a, false, bb1, (short)0, acc1, false, false);
        }

        // epilogue: FiLM scale/bias, relu, weighted channel-partial
        float pa0 = 0.f, pa1 = 0.f;
        #pragma unroll
        for (int r = 0; r < 8; ++r) {
            int ch = i0 * 16 + r + hi * 8;
            float y0 = fmaxf(SC[ch] * (acc0[r] + D0[ch]) + TB[ch], 0.f);
            float y1 = fmaxf(SC[ch] * (acc1[r] + D0[ch]) + TB[ch], 0.f);
            pa0 += VV[ch] * y0;
            pa1 += VV[ch] * y1;
        }
        pa0 += __shfl_down(pa0, 16, 32);      // combine row-halves (same column)
        pa1 += __shfl_down(pa1, 16, 32);
        if (hi == 0) {
            part[i0][j0 * 16 + lo] = pa0;     // each (i,j) tile unique -> no collision
            part[i0][j1 * 16 + lo] = pa1;
        }
        __syncthreads();
        if (t < 64) {
            float v = part[0][t] + part[1][t] + part[2][t] + part[3][t] + sb;
            if (br == 0) v = v / (1.f + fabsf(v));   // soft_sign for lv branch
            orow[n0 + t] = v;
        }
        __syncthreads();
    }
}

// ---------------- K5: final elementwise ----------------
__global__ void k5_final(const float* __restrict__ p, float* __restrict__ outb) {
    int i = blockIdx.x * 256 + threadIdx.x;
    if (i >= M_) return;
    int b = i >> 14, n = i & (N_ - 1);
    long base = (long)b * 3 * N_ + n;
    float lvw = outb[(long)2 * OUTS + base];
    float muw = outb[(long)OUTS + base];
    float p0 = p[base], p1 = p[base + N_], p2 = p[base + 2 * N_];
    float s0 = sqrtf(1e-6f + expf(lvw));
    const float s12 = sqrtf(1.f + 1e-6f);     // logvar==0, mu==0 for ch 1,2
    outb[base]            = s0 * p0 + muw;
    outb[base + N_]       = s12 * p1;
    outb[base + 2 * N_]   = s12 * p2;
    outb[(long)OUTS + base + N_]         = 0.f;
    outb[(long)OUTS + base + 2 * N_]     = 0.f;
    outb[(long)2 * OUTS + base + N_]     = 0.f;
    outb[(long)2 * OUTS + base + 2 * N_] = 0.f;
}

extern "C" void kernel_launch(void* const* d_in, const int* in_sizes, int n_in,
                              void* d_out, int out_size, void* d_ws, size_t ws_size,
                              hipStream_t stream) {
    (void)in_sizes; (void)n_in; (void)out_size; (void)ws_size;
    const float* p = (const float*)d_in[0];
    const float* g = (const float*)d_in[1];
    float* ws   = (float*)d_ws;
    float* outb = (float*)d_out;
    auto in = [&](int i) { return (const float*)d_in[i]; };

    k0_moments<<<NWG0, 256, 0, stream>>>(p, ws);
    k1_prep<<<1, 256, 0, stream>>>(ws, in(2), in(3), in(4), in(18), in(19), in(20));
    for (int br = 0; br < 2; ++br) {
        int base = 2 + br * 16;
        k2_gram<<<NWG2, 256, 0, stream>>>(p, ws, br);
        k2b_reduce<<<17, 256, 0, stream>>>(ws);
        k2c_bn2<<<1, 64, 0, stream>>>(ws, in(base + 3), br);                 // sd1_W
        k3_film<<<1, 256, 0, stream>>>(g, in(base + 4), in(base + 5), in(base + 6),
                                       in(base + 7), in(base + 8),
                                       ws + OFF_FILM + br * 8192, 1);        // cw -> scale
        k3_film<<<1, 256, 0, stream>>>(g, in(base + 9), in(base + 10), in(base + 11),
                                       in(base + 12), in(base + 13),
                                       ws + OFF_FILM + br * 8192 + 4096, 0); // cb -> bias
        k4_main<<<2048, 256, 0, stream>>>(p, ws, in(base + 14), in(base + 15), outb, br);
    }
    k5_final<<<M_ / 256, 256, 0, stream>>>(p, outb);
}